// BoundaryUnit_9835475108249
// MI455X (gfx1250) — compile-verified
//
#include <hip/hip_runtime.h>
#include <math.h>

typedef float v2f __attribute__((ext_vector_type(2)));
typedef float v8f __attribute__((ext_vector_type(8)));

constexpr int Bc = 4, Nc = 128, Lc = 20, Dc = 512;

// ---------------------------------------------------------------------------
// C = scale * (A @ W^T) + bias     A:[M,K] row-major, W:[N,K] row-major.
// One wave per 16x16 output tile; fp32 WMMA, K stepped by 4.
// A fragment (16x4): lane<16 holds M=lane, K=k..k+1; lane>=16 holds K=k+2..k+3
// B fragment (4x16): lane<16 holds N=lane, K=k..k+1; lane>=16 holds K=k+2..k+3
// D (16x16 f32): VGPR r, lanes 0-15 -> M=r,N=lane ; lanes 16-31 -> M=r+8.
// ---------------------------------------------------------------------------
__global__ void gemm_nt_wmma(const float* __restrict__ A,
                             const float* __restrict__ W,
                             const float* __restrict__ bias,
                             float* __restrict__ C,
                             int M, int N, int K,
                             long sA, long sW, long sC,
                             float scale) {
  const int batch = blockIdx.z;
  A += (long)batch * sA;
  W += (long)batch * sW;
  C += (long)batch * sC;

  const int tiles_n = N >> 4;
  const int tiles_m = M >> 4;
  const int tile = blockIdx.x * (blockDim.x >> 5) + (threadIdx.x >> 5);
  if (tile >= tiles_m * tiles_n) return;          // uniform per-wave exit
  const int tm = tile / tiles_n;
  const int tn = tile - tm * tiles_n;

  const int lane = threadIdx.x & 31;
  const int half = lane >> 4;                     // which K half of the step
  const int lidx = lane & 15;                     // M (for A) / N (for B)

  const float* a_ptr = A + (long)(tm * 16 + lidx) * K + 2 * half;
  const float* w_ptr = W + (long)(tn * 16 + lidx) * K + 2 * half;

  v8f acc = {0.f, 0.f, 0.f, 0.f, 0.f, 0.f, 0.f, 0.f};
  #pragma unroll 8
  for (int k = 0; k < K; k += 4) {
    v2f a = { a_ptr[k], a_ptr[k + 1] };
    v2f b = { w_ptr[k], w_ptr[k + 1] };
    acc = __builtin_amdgcn_wmma_f32_16x16x4_f32(
        /*neg_a=*/false, a, /*neg_b=*/false, b,
        /*c_mod=*/(short)0, acc, /*reuse_a=*/false, /*reuse_b=*/false);
  }

  const int n = tn * 16 + lidx;
  const float bv = bias ? bias[n] : 0.f;
  const int m0 = tm * 16 + half * 8;
  #pragma unroll
  for (int r = 0; r < 8; ++r)
    C[(long)(m0 + r) * N + n] = acc[r] * scale + bv;
}

// ---------------------------------------------------------------------------
// C = A @ Bm     A:[M,K] row-major, Bm:[K,N] row-major (no transpose).
// ---------------------------------------------------------------------------
__global__ void gemm_nn_wmma(const float* __restrict__ A,
                             const float* __restrict__ Bm,
                             float* __restrict__ C,
                             int M, int N, int K,
                             long sA, long sB, long sC) {
  const int batch = blockIdx.z;
  A  += (long)batch * sA;
  Bm += (long)batch * sB;
  C  += (long)batch * sC;

  const int tiles_n = N >> 4;
  const int tiles_m = M >> 4;
  const int tile = blockIdx.x * (blockDim.x >> 5) + (threadIdx.x >> 5);
  if (tile >= tiles_m * tiles_n) return;
  const int tm = tile / tiles_n;
  const int tn = tile - tm * tiles_n;

  const int lane = threadIdx.x & 31;
  const int half = lane >> 4;
  const int lidx = lane & 15;

  const float* a_ptr = A + (long)(tm * 16 + lidx) * K + 2 * half;
  const float* b_ptr = Bm + (long)(2 * half) * N + tn * 16 + lidx;

  v8f acc = {0.f, 0.f, 0.f, 0.f, 0.f, 0.f, 0.f, 0.f};
  #pragma unroll 8
  for (int k = 0; k < K; k += 4) {
    v2f a = { a_ptr[k], a_ptr[k + 1] };
    v2f b = { b_ptr[(long)k * N], b_ptr[(long)(k + 1) * N] };
    acc = __builtin_amdgcn_wmma_f32_16x16x4_f32(
        false, a, false, b, (short)0, acc, false, false);
  }

  const int n = tn * 16 + lidx;
  const int m0 = tm * 16 + half * 8;
  #pragma unroll
  for (int r = 0; r < 8; ++r)
    C[(long)(m0 + r) * N + n] = acc[r];
}

// ---------------------------------------------------------------------------
// Per-(b,n): softmax over L=20 word scores, f_baq = attn @ f_w,
// f_bq = f_b * (f_baq + f_s).  256 threads, each owns d=t and d=t+256.
// ---------------------------------------------------------------------------
__global__ void attn_gate_kernel(const float* __restrict__ q,
                                 const float* __restrict__ kmat,
                                 const float* __restrict__ f_w,
                                 const float* __restrict__ f_b,
                                 const float* __restrict__ f_s,
                                 float* __restrict__ f_bq) {
  const int bn = blockIdx.x;            // b*Nc + n
  const int b = bn >> 7;
  const int t = threadIdx.x;            // 0..255
  __shared__ float s_part[Lc][9];
  __shared__ float s_attn[Lc];

  const float q0 = q[(long)bn * Dc + t];
  const float q1 = q[(long)bn * Dc + t + 256];

  float p[Lc];
  #pragma unroll
  for (int l = 0; l < Lc; ++l) {
    const float* kr = kmat + (long)(b * Lc + l) * Dc;
    p[l] = q0 * kr[t] + q1 * kr[t + 256];
  }

  const int lane = t & 31, wave = t >> 5;
  #pragma unroll
  for (int l = 0; l < Lc; ++l) {
    float v = p[l];
    for (int off = 16; off; off >>= 1) v += __shfl_xor(v, off, 32);
    if (lane == 0) s_part[l][wave] = v;
  }
  __syncthreads();

  if (t == 0) {
    const float scale = 0.044194173824159216f;   // 1/sqrt(512)
    float sc[Lc];
    float mx = -1e30f;
    for (int l = 0; l < Lc; ++l) {
      float s = 0.f;
      for (int w = 0; w < 8; ++w) s += s_part[l][w];
      s *= scale;
      sc[l] = s;
      mx = fmaxf(mx, s);
    }
    float sum = 0.f;
    for (int l = 0; l < Lc; ++l) { sc[l] = __expf(sc[l] - mx); sum += sc[l]; }
    const float inv = 1.f / sum;
    for (int l = 0; l < Lc; ++l) s_attn[l] = sc[l] * inv;
  }
  __syncthreads();

  float fba0 = 0.f, fba1 = 0.f;
  #pragma unroll
  for (int l = 0; l < Lc; ++l) {
    const float* wr = f_w + (long)(b * Lc + l) * Dc;
    const float a = s_attn[l];
    fba0 += a * wr[t];
    fba1 += a * wr[t + 256];
  }
  const long i0 = (long)bn * Dc + t, i1 = i0 + 256;
  f_bq[i0] = f_b[i0] * (fba0 + f_s[(long)b * Dc + t]);
  f_bq[i1] = f_b[i1] * (fba1 + f_s[(long)b * Dc + t + 256]);
}

// ---------------------------------------------------------------------------
// In-place row softmax over 128 columns. One wave per row, 4 cols/lane.
// ---------------------------------------------------------------------------
__global__ void softmax128_kernel(float* __restrict__ S) {
  const int row = blockIdx.x * 8 + (threadIdx.x >> 5);
  const int lane = threadIdx.x & 31;
  float* r = S + (long)row * Nc;

  float v[4];
  float mx = -1e30f;
  #pragma unroll
  for (int i = 0; i < 4; ++i) { v[i] = r[lane + 32 * i]; mx = fmaxf(mx, v[i]); }
  for (int off = 16; off; off >>= 1) mx = fmaxf(mx, __shfl_xor(mx, off, 32));
  float sum = 0.f;
  #pragma unroll
  for (int i = 0; i < 4; ++i) { v[i] = __expf(v[i] - mx); sum += v[i]; }
  for (int off = 16; off; off >>= 1) sum += __shfl_xor(sum, off, 32);
  const float inv = 1.f / sum;
  #pragma unroll
  for (int i = 0; i < 4; ++i) r[lane + 32 * i] = v[i] * inv;
}

// ---------------------------------------------------------------------------
// out[b,j,:] = f_bb + f_b + sum_i A_b[b,i,j] * sigmoid(f_m*f_s)*f_m
// One block per (b,j); 256 threads cover d = t and t+256 (full 2KB row / i).
// f_m is a single-use 134 MB stream -> non-temporal loads + ahead prefetch,
// unrolled x4 so ~8 loads are in flight per wave per waitcnt window.
// ---------------------------------------------------------------------------
__global__ void moment_out_kernel(const float* __restrict__ A_b,
                                  const float* __restrict__ f_m,
                                  const float* __restrict__ f_s,
                                  const float* __restrict__ f_bb,
                                  const float* __restrict__ f_b,
                                  float* __restrict__ out) {
  const int bj = blockIdx.x;
  const int b = bj >> 7, j = bj & 127;
  const int t = threadIdx.x;
  __shared__ float sA[Nc];
  if (t < Nc) sA[t] = A_b[((long)(b * Nc + t)) * Nc + j];
  __syncthreads();

  const float s0 = f_s[(long)b * Dc + t];
  const float s1 = f_s[(long)b * Dc + t + 256];
  const float* fm = f_m + (long)b * Nc * Nc * Dc + (long)j * Dc;
  const long istride = (long)Nc * Dc;

  float acc0 = 0.f, acc1 = 0.f;
  #pragma unroll 4
  for (int i = 0; i < Nc; ++i) {
    const float* row = fm + (long)i * istride;
    if (i + 4 < Nc) {
      __builtin_prefetch(row + 4 * istride + t, 0, 1);       // global_prefetch_b8
      __builtin_prefetch(row + 4 * istride + t + 256, 0, 1);
    }
    const float a = sA[i];
    const float m0 = __builtin_nontemporal_load(row + t);
    const float m1 = __builtin_nontemporal_load(row + t + 256);
    const float g0 = 1.f / (1.f + __expf(-m0 * s0));
    const float g1 = 1.f / (1.f + __expf(-m1 * s1));
    acc0 += a * g0 * m0;
    acc1 += a * g1 * m1;
  }
  const long o0 = (long)bj * Dc + t, o1 = o0 + 256;
  out[o0] = f_bb[o0] + f_b[o0] + acc0;
  out[o1] = f_bb[o1] + f_b[o1] + acc1;
}

// ---------------------------------------------------------------------------
extern "C" void kernel_launch(void* const* d_in, const int* in_sizes, int n_in,
                              void* d_out, int out_size, void* d_ws, size_t ws_size,
                              hipStream_t stream) {
  const float* f_b = (const float*)d_in[0];
  const float* f_w = (const float*)d_in[1];
  const float* f_s = (const float*)d_in[2];
  const float* f_m = (const float*)d_in[3];
  const float* Wq  = (const float*)d_in[4];
  const float* bq  = (const float*)d_in[5];
  const float* Wk  = (const float*)d_in[6];
  const float* bk  = (const float*)d_in[7];
  float* out = (float*)d_out;

  float* ws   = (float*)d_ws;
  float* q    = ws;                              // 512*512
  float* kmat = q + (long)Bc * Nc * Dc;          // 80*512
  float* f_bq = kmat + (long)Bc * Lc * Dc;       // 512*512
  float* S    = f_bq + (long)Bc * Nc * Dc;       // 4*128*128
  float* f_bb = S + (long)Bc * Nc * Nc;          // 512*512

  const float scale = 0.044194173824159216f;     // 1/sqrt(512)
  const int WPB = 4;                             // waves (tiles) per block

  // 1) q = f_b @ Wq^T + bq    (M=BN=512, N=512, K=512)
  {
    const int tiles = (Bc * Nc / 16) * (Dc / 16);
    gemm_nt_wmma<<<dim3((tiles + WPB - 1) / WPB, 1, 1), 32 * WPB, 0, stream>>>(
        f_b, Wq, bq, q, Bc * Nc, Dc, Dc, 0, 0, 0, 1.0f);
  }
  // 2) k = f_w @ Wk^T + bk    (M=BL=80, N=512, K=512)
  {
    const int tiles = (Bc * Lc / 16) * (Dc / 16);
    gemm_nt_wmma<<<dim3((tiles + WPB - 1) / WPB, 1, 1), 32 * WPB, 0, stream>>>(
        f_w, Wk, bk, kmat, Bc * Lc, Dc, Dc, 0, 0, 0, 1.0f);
  }
  // 3) cross-attn softmax + gating -> f_bq
  attn_gate_kernel<<<Bc * Nc, 256, 0, stream>>>(q, kmat, f_w, f_b, f_s, f_bq);
  // 4) S = scale * f_bq @ f_bq^T per batch   (M=N=128, K=512)
  {
    const int tiles = (Nc / 16) * (Nc / 16);
    gemm_nt_wmma<<<dim3((tiles + WPB - 1) / WPB, 1, Bc), 32 * WPB, 0, stream>>>(
        f_bq, f_bq, nullptr, S, Nc, Nc, Dc,
        (long)Nc * Dc, (long)Nc * Dc, (long)Nc * Nc, scale);
  }
  // 5) row softmax -> A_b (in place)
  softmax128_kernel<<<(Bc * Nc) / 8, 256, 0, stream>>>(S);
  // 6) f_bb = A_b @ f_b per batch   (M=128, N=512, K=128)
  {
    const int tiles = (Nc / 16) * (Dc / 16);
    gemm_nn_wmma<<<dim3((tiles + WPB - 1) / WPB, 1, Bc), 32 * WPB, 0, stream>>>(
        S, f_b, f_bb, Nc, Dc, Nc,
        (long)Nc * Nc, (long)Nc * Dc, (long)Nc * Dc);
  }
  // 7) stream f_m (134 MB -> BW bound), fuse final sum
  moment_out_kernel<<<Bc * Nc, 256, 0, stream>>>(S, f_m, f_s, f_bb, f_b, out);
}